// SO2EquivariantGraphAttention_84859963834913
// MI455X (gfx1250) — compile-verified
//
#include <hip/hip_runtime.h>
#include <math.h>

// ---------------------------------------------------------------------------
// MI455X (gfx1250) implementation of SO2EquivariantGraphAttention.
// Heavy per-edge GEMMs run on V_WMMA_F32_16X16X4_F32 (full fp32 precision,
// matching the fp32 reference). Edges are tiled 16 per wave (wave32).
// Weights are repacked once per launch into WMMA-fragment-major order so each
// B fragment is a single coalesced global_load_b64 (256B per wave per step).
// ---------------------------------------------------------------------------

typedef float v2f __attribute__((ext_vector_type(2)));
typedef float v8f __attribute__((ext_vector_type(8)));

__device__ __forceinline__ v8f wmma_f32(v2f a, v2f b, v8f c) {
  // D = A(16x4) * B(4x16) + C(16x16), fp32
  return __builtin_amdgcn_wmma_f32_16x16x4_f32(
      /*neg_a=*/false, a, /*neg_b=*/false, b,
      /*c_mod=*/(short)0, c, /*reuse_a=*/false, /*reuse_b=*/false);
}

// A fragment: 16 rows x 4 K from row-major (lda = leading dim).
// lanes 0-15 : row = lane, v0=K0, v1=K1 ; lanes 16-31: row = lane-16, v0=K2, v1=K3
__device__ __forceinline__ v2f frag_a(const float* A, int lda, int k0, int lane) {
  int row = lane & 15;
  int kk  = k0 + ((lane >> 4) << 1);
  v2f a;
  a.x = A[row * lda + kk];
  a.y = A[row * lda + kk + 1];
  return a;
}

// B fragment from packed (fragment-major) weights: one b64 load per step.
// Wp index: ((nt * K/4 + ks) * 32 + lane) v2f elements.
__device__ __forceinline__ v2f frag_b_pk(const float* Wp, int k4, int nt, int ks, int lane) {
  const v2f* p = (const v2f*)Wp;
  return p[(size_t)(nt * k4 + ks) * 32 + lane];
}

__device__ __forceinline__ float dev_silu(float x) { return x / (1.f + __expf(-x)); }
__device__ __forceinline__ float dev_sigmoid(float x) { return 1.f / (1.f + __expf(-x)); }

__device__ __forceinline__ void atomicMaxF(float* addr, float val) {
  int* ai = (int*)addr;
  int old = __float_as_int(*addr);
  while (__int_as_float(old) < val) {
    int prev = atomicCAS(ai, old, __float_as_int(val));
    if (prev == old) break;
    old = prev;
  }
}

// ---------------------------------------------------------------------------
// K0a: repack row-major W(K,N) into WMMA B-fragment-major layout.
// ---------------------------------------------------------------------------
__global__ void repack_b_kernel(const float* __restrict__ W, float* __restrict__ Wp,
                                int K, int N) {
  int t = blockIdx.x * blockDim.x + threadIdx.x;
  int k4 = K >> 2;
  int total = k4 * (N >> 4) * 32;  // number of v2f elements
  if (t >= total) return;
  int lane = t & 31;
  int rest = t >> 5;
  int ks = rest % k4;
  int nt = rest / k4;
  int col = (nt << 4) + (lane & 15);
  int kk = (ks << 2) + ((lane >> 4) << 1);
  float2 v;
  v.x = W[(size_t)kk * N + col];
  v.y = W[(size_t)(kk + 1) * N + col];
  ((float2*)Wp)[t] = v;
}

// ---------------------------------------------------------------------------
// K0b: init accumulators (amax=-inf, den=0, outacc=0)
// ---------------------------------------------------------------------------
__global__ void init_kernel(float* amax, float* den, float* outacc, int N) {
  int i = blockIdx.x * blockDim.x + threadIdx.x;
  if (i < N * 8) { amax[i] = -3.0e38f; den[i] = 0.f; }
  if (i < N * 576) outacc[i] = 0.f;
}

// ---------------------------------------------------------------------------
// K1: msg = wigner @ concat(nf[snd], nf[rcv])   (E,9,128)
// ---------------------------------------------------------------------------
__global__ void wigner_msg_kernel(const float* __restrict__ nf,
                                  const float* __restrict__ wig,
                                  const int* __restrict__ snd,
                                  const int* __restrict__ rcv,
                                  float* __restrict__ msg1, int E) {
  __shared__ float sin_[9 * 128];
  __shared__ float sw[81];
  int e = blockIdx.x;
  if (e >= E) return;
  int tid = threadIdx.x;
  int s = snd[e], r = rcv[e];
  for (int o = tid; o < 1152; o += blockDim.x) {
    int i = o >> 7, c = o & 127;
    sin_[o] = (c < 64) ? nf[(size_t)s * 576 + i * 64 + c]
                       : nf[(size_t)r * 576 + i * 64 + (c - 64)];
  }
  if (tid < 81) sw[tid] = wig[(size_t)e * 81 + tid];
  __syncthreads();
  for (int o = tid; o < 1152; o += blockDim.x) {
    int i = o >> 7, c = o & 127;
    float acc = 0.f;
#pragma unroll
    for (int k = 0; k < 9; ++k) acc += sw[i * 9 + k] * sin_[k * 128 + c];
    msg1[(size_t)e * 1152 + o] = acc;
  }
}

// ---------------------------------------------------------------------------
// K2: radial MLP. Block = 4 waves, one 16-edge tile, WMMA f32.
// ---------------------------------------------------------------------------
__global__ void radial_kernel(const float* __restrict__ ee, const float* __restrict__ se,
                              const int* __restrict__ spec, const int* __restrict__ snd,
                              const int* __restrict__ rcv,
                              const float* __restrict__ rw1p, const float* __restrict__ rb1,
                              const float* __restrict__ rw2p, const float* __restrict__ rb2,
                              const float* __restrict__ rw3p, const float* __restrict__ rb3,
                              float* __restrict__ rad, int E) {
  __shared__ float eld[16 * 192];
  __shared__ float h1[16 * 64];
  __shared__ float h2[16 * 64];
  int e0 = blockIdx.x * 16;
  int tid = threadIdx.x, lane = tid & 31, wave = tid >> 5;

  for (int idx = tid; idx < 16 * 192; idx += blockDim.x) {
    int row = idx / 192, t = idx - row * 192;
    int e = e0 + row;
    float v = 0.f;
    if (e < E) {
      if (t < 64)       v = ee[(size_t)e * 64 + t];
      else if (t < 128) v = se[(size_t)spec[snd[e]] * 128 + (t - 64)];
      else              v = se[(size_t)spec[rcv[e]] * 128 + 64 + (t - 128)];
    }
    eld[idx] = v;
  }
  __syncthreads();

  {  // layer 1: (16x192)@(192x64), one 16-col tile per wave; K/4 = 48
    v8f acc = {};
    for (int ks = 0; ks < 48; ++ks)
      acc = wmma_f32(frag_a(eld, 192, ks << 2, lane), frag_b_pk(rw1p, 48, wave, ks, lane), acc);
    int col = (wave << 4) + (lane & 15), mb = (lane >> 4) << 3;
#pragma unroll
    for (int j = 0; j < 8; ++j) h1[(mb + j) * 64 + col] = dev_silu(acc[j] + rb1[col]);
  }
  __syncthreads();
  {  // layer 2: (16x64)@(64x64); K/4 = 16
    v8f acc = {};
    for (int ks = 0; ks < 16; ++ks)
      acc = wmma_f32(frag_a(h1, 64, ks << 2, lane), frag_b_pk(rw2p, 16, wave, ks, lane), acc);
    int col = (wave << 4) + (lane & 15), mb = (lane >> 4) << 3;
#pragma unroll
    for (int j = 0; j < 8; ++j) h2[(mb + j) * 64 + col] = dev_silu(acc[j] + rb2[col]);
  }
  __syncthreads();
  // layer 3: (16x64)@(64x768), 48 col tiles, 12 per wave
  for (int nt = wave; nt < 48; nt += 4) {
    __builtin_prefetch(&rw3p[(size_t)nt * 16 * 64], 0, 1);
    v8f acc = {};
    for (int ks = 0; ks < 16; ++ks)
      acc = wmma_f32(frag_a(h2, 64, ks << 2, lane), frag_b_pk(rw3p, 16, nt, ks, lane), acc);
    int col = (nt << 4) + (lane & 15), mb = (lane >> 4) << 3;
#pragma unroll
    for (int j = 0; j < 8; ++j) {
      int e = e0 + mb + j;
      if (e < E) rad[(size_t)e * 768 + col] = acc[j] + rb3[col];
    }
  }
}

// ---------------------------------------------------------------------------
// K3: generic SO2 conv (conv1 with rad/extra, conv2 without).
// Block = 4 waves, one 16-edge tile.  M0={0,2,6} N1={1,5} P1={3,7} N2={4} P2={8}
// ---------------------------------------------------------------------------
__global__ void so2conv_kernel(const float* __restrict__ X,    // (E,9,cin)
                               const float* __restrict__ rad,  // (E,6*cin) or null
                               const float* __restrict__ w0p, const float* __restrict__ b0,
                               const float* __restrict__ w1p, const float* __restrict__ w2p,
                               float* __restrict__ Y,          // (E,9,cout)
                               float* __restrict__ extra,      // (E,n_extra) or null
                               int E, int cin, int cout, int n_extra) {
  __shared__ float lds[8192];  // 32 KB: max(16*3cin, 2*16*2cin)
  int e0 = blockIdx.x * 16;
  int tid = threadIdx.x, lane = tid & 31, wave = tid >> 5;
  const int NW = blockDim.x >> 5;
  const int rs = 6 * cin;

  // ---- phase 0: m=0 block ------------------------------------------------
  const int K0 = 3 * cin;
  for (int idx = tid; idx < 16 * K0; idx += blockDim.x) {
    int row = idx / K0, t = idx - row * K0;
    int j = t / cin, c = t - j * cin;
    int comp = (j == 0) ? 0 : ((j == 1) ? 2 : 6);
    int e = e0 + row;
    float v = 0.f;
    if (e < E) {
      v = X[(size_t)e * 9 * cin + comp * cin + c];
      if (rad) v *= rad[(size_t)e * rs + t];
    }
    lds[row * K0 + t] = v;
  }
  __syncthreads();
  const int N0 = 3 * cout + n_extra;
  const int k40 = K0 >> 2;
  for (int nt = wave; nt * 16 < N0; nt += NW) {
    __builtin_prefetch(&w0p[(size_t)nt * k40 * 64], 0, 1);
    v8f acc = {};
    for (int ks = 0; ks < k40; ++ks)
      acc = wmma_f32(frag_a(lds, K0, ks << 2, lane), frag_b_pk(w0p, k40, nt, ks, lane), acc);
    int col = (nt << 4) + (lane & 15), mb = (lane >> 4) << 3;
#pragma unroll
    for (int j = 0; j < 8; ++j) {
      int e = e0 + mb + j;
      if (e >= E) continue;
      float val = acc[j] + b0[col];
      if (col < 3 * cout) {
        int q = col / cout;
        int comp = (q == 0) ? 0 : ((q == 1) ? 2 : 6);
        Y[(size_t)e * 9 * cout + comp * cout + (col - q * cout)] = val;
      } else if (extra) {
        extra[(size_t)e * n_extra + (col - 3 * cout)] = val;
      }
    }
  }
  __syncthreads();

  // ---- phase 1: m=1 block ------------------------------------------------
  const int K1 = 2 * cin;
  float* xn = lds;
  float* xp = lds + 16 * K1;
  for (int idx = tid; idx < 2 * 16 * K1; idx += blockDim.x) {
    int half = idx / (16 * K1);
    int rem = idx - half * 16 * K1;
    int row = rem / K1, t = rem - row * K1;
    int j = t / cin, c = t - j * cin;
    int comp = half ? (j ? 7 : 3) : (j ? 5 : 1);
    int e = e0 + row;
    float v = 0.f;
    if (e < E) {
      v = X[(size_t)e * 9 * cin + comp * cin + c];
      if (rad) v *= rad[(size_t)e * rs + 3 * cin + t];
    }
    (half ? xp : xn)[row * K1 + t] = v;
  }
  __syncthreads();
  {
    const int h = 2 * cout;       // half of w1's output width
    const int k41 = K1 >> 2;
    const int hti = h >> 4;       // tile offset of imag half
    for (int tp = wave; tp * 16 < h; tp += NW) {
      v8f anr = {}, apr = {}, ani = {}, api = {};
      for (int ks = 0; ks < k41; ++ks) {
        v2f fan = frag_a(xn, K1, ks << 2, lane);
        v2f fap = frag_a(xp, K1, ks << 2, lane);
        v2f fbr = frag_b_pk(w1p, k41, tp, ks, lane);
        v2f fbi = frag_b_pk(w1p, k41, tp + hti, ks, lane);
        anr = wmma_f32(fan, fbr, anr);
        apr = wmma_f32(fap, fbr, apr);
        ani = wmma_f32(fan, fbi, ani);
        api = wmma_f32(fap, fbi, api);
      }
      int col = (tp << 4) + (lane & 15), mb = (lane >> 4) << 3;
      int q = col / cout, ch = col - q * cout;
      int compn = q ? 5 : 1, compp = q ? 7 : 3;
#pragma unroll
      for (int j = 0; j < 8; ++j) {
        int e = e0 + mb + j;
        if (e >= E) continue;
        Y[(size_t)e * 9 * cout + compn * cout + ch] = anr[j] - api[j];
        Y[(size_t)e * 9 * cout + compp * cout + ch] = apr[j] + ani[j];
      }
    }
  }
  __syncthreads();

  // ---- phase 2: m=2 block ------------------------------------------------
  const int K2 = cin;
  xn = lds;
  xp = lds + 16 * K2;
  for (int idx = tid; idx < 2 * 16 * K2; idx += blockDim.x) {
    int half = idx / (16 * K2);
    int rem = idx - half * 16 * K2;
    int row = rem / K2, c = rem - row * K2;
    int comp = half ? 8 : 4;
    int e = e0 + row;
    float v = 0.f;
    if (e < E) {
      v = X[(size_t)e * 9 * cin + comp * cin + c];
      if (rad) v *= rad[(size_t)e * rs + 5 * cin + c];
    }
    (half ? xp : xn)[row * K2 + c] = v;
  }
  __syncthreads();
  {
    const int h = cout;
    const int k42 = K2 >> 2;
    const int hti = h >> 4;
    for (int tp = wave; tp * 16 < h; tp += NW) {
      v8f anr = {}, apr = {}, ani = {}, api = {};
      for (int ks = 0; ks < k42; ++ks) {
        v2f fan = frag_a(xn, K2, ks << 2, lane);
        v2f fap = frag_a(xp, K2, ks << 2, lane);
        v2f fbr = frag_b_pk(w2p, k42, tp, ks, lane);
        v2f fbi = frag_b_pk(w2p, k42, tp + hti, ks, lane);
        anr = wmma_f32(fan, fbr, anr);
        apr = wmma_f32(fap, fbr, apr);
        ani = wmma_f32(fan, fbi, ani);
        api = wmma_f32(fap, fbi, api);
      }
      int col = (tp << 4) + (lane & 15), mb = (lane >> 4) << 3;
#pragma unroll
      for (int j = 0; j < 8; ++j) {
        int e = e0 + mb + j;
        if (e >= E) continue;
        Y[(size_t)e * 9 * cout + 4 * cout + col] = anr[j] - api[j];
        Y[(size_t)e * 9 * cout + 8 * cout + col] = apr[j] + ani[j];
      }
    }
  }
}

// ---------------------------------------------------------------------------
// K4: grid nonlinearity + gate:  m2in = concat(silu(gate), (fg @ silu(tg @ y))[1:])
// ---------------------------------------------------------------------------
__global__ void grid_gate_kernel(const float* __restrict__ ybuf,
                                 const float* __restrict__ extrabuf,
                                 const float* __restrict__ tg,
                                 const float* __restrict__ fg,
                                 float* __restrict__ m2in, int E) {
  __shared__ float stg[144], sfg[144];
  int tid = threadIdx.x;
  if (tid < 144) { stg[tid] = tg[tid]; sfg[tid] = fg[tid]; }
  __syncthreads();
  int t = blockIdx.x * blockDim.x + tid;
  int e = t >> 6, c = t & 63;
  if (e >= E) return;
  float yv[9];
#pragma unroll
  for (int i = 0; i < 9; ++i) yv[i] = ybuf[(size_t)e * 576 + i * 64 + c];
  float gv[16];
#pragma unroll
  for (int g = 0; g < 16; ++g) {
    float s = 0.f;
#pragma unroll
    for (int i = 0; i < 9; ++i) s += stg[g * 9 + i] * yv[i];
    gv[g] = dev_silu(s);
  }
#pragma unroll
  for (int i = 1; i < 9; ++i) {
    float v = 0.f;
#pragma unroll
    for (int g = 0; g < 16; ++g) v += sfg[i * 16 + g] * gv[g];
    m2in[(size_t)e * 576 + i * 64 + c] = v;
  }
  m2in[(size_t)e * 576 + c] = dev_silu(extrabuf[(size_t)e * 576 + 512 + c]);
}

// ---------------------------------------------------------------------------
// K5: per-head layernorm + swish-mix + alpha dot; wave32 per edge.
// ---------------------------------------------------------------------------
__global__ void alpha_kernel(const float* __restrict__ extrabuf,
                             const float* __restrict__ ln_s, const float* __restrict__ ln_b,
                             const float* __restrict__ ad, const int* __restrict__ rcv,
                             float* __restrict__ alphabuf, float* __restrict__ amax, int E) {
  int lane = threadIdx.x & 31;
  int e = blockIdx.x * (blockDim.x >> 5) + (threadIdx.x >> 5);
  if (e >= E) return;
  int r = rcv[e];
  const float* sa = extrabuf + (size_t)e * 576;
  float ls0 = ln_s[lane], ls1 = ln_s[lane + 32];
  float lb0 = ln_b[lane], lb1 = ln_b[lane + 32];
  for (int h = 0; h < 8; ++h) {
    float x0 = sa[h * 64 + lane];
    float x1 = sa[h * 64 + 32 + lane];
    float s = x0 + x1;
    for (int o = 16; o > 0; o >>= 1) s += __shfl_xor(s, o, 32);
    float mu = s * (1.f / 64.f);
    float d0 = x0 - mu, d1 = x1 - mu;
    float v = d0 * d0 + d1 * d1;
    for (int o = 16; o > 0; o >>= 1) v += __shfl_xor(v, o, 32);
    float rstd = rsqrtf(v * (1.f / 64.f) + 1e-6f);
    float a0 = d0 * rstd * ls0 + lb0;
    float a1 = d1 * rstd * ls1 + lb1;
    float t0 = 0.2f * a0 + 0.8f * a0 * dev_sigmoid(a0);
    float t1 = 0.2f * a1 + 0.8f * a1 * dev_sigmoid(a1);
    float p = t0 * ad[lane * 8 + h] + t1 * ad[(lane + 32) * 8 + h];
    for (int o = 16; o > 0; o >>= 1) p += __shfl_xor(p, o, 32);
    if (lane == 0) {
      alphabuf[(size_t)e * 8 + h] = p;
      atomicMaxF(&amax[r * 8 + h], p);
    }
  }
}

// ---------------------------------------------------------------------------
// K6: ex = exp(alpha - amax[r]);  den[r] += ex
// ---------------------------------------------------------------------------
__global__ void exp_den_kernel(float* __restrict__ alphabuf, const float* __restrict__ amax,
                               float* __restrict__ den, const int* __restrict__ rcv, int E) {
  int i = blockIdx.x * blockDim.x + threadIdx.x;
  if (i >= E * 8) return;
  int e = i >> 3, h = i & 7;
  int r = rcv[e];
  float ex = __expf(alphabuf[i] - amax[r * 8 + h]);
  alphabuf[i] = ex;
  atomicAdd(&den[r * 8 + h], ex);
}

// ---------------------------------------------------------------------------
// K7: alpha-scale msg2, wigner_inv rotate, segment scatter-add to outacc
// ---------------------------------------------------------------------------
__global__ void combine_kernel(const float* __restrict__ m2out, const float* __restrict__ winv,
                               const float* __restrict__ alphabuf, const float* __restrict__ den,
                               const int* __restrict__ rcv, float* __restrict__ outacc, int E) {
  __shared__ float sw[81];
  __shared__ float sm[576];
  __shared__ float an[8];
  int e = blockIdx.x;
  if (e >= E) return;
  int tid = threadIdx.x;
  int r = rcv[e];
  if (tid < 81) sw[tid] = winv[(size_t)e * 81 + tid];
  if (tid >= 96 && tid < 104) {
    int h = tid - 96;
    an[h] = alphabuf[(size_t)e * 8 + h] / (den[r * 8 + h] + 1e-16f);
  }
  __syncthreads();
  for (int o = tid; o < 576; o += blockDim.x)
    sm[o] = m2out[(size_t)e * 576 + o] * an[(o & 63) >> 3];
  __syncthreads();
  for (int o = tid; o < 576; o += blockDim.x) {
    int i = o >> 6, c = o & 63;
    float v = 0.f;
#pragma unroll
    for (int k = 0; k < 9; ++k) v += sw[i * 9 + k] * sm[k * 64 + c];
    atomicAdd(&outacc[(size_t)r * 576 + o], v);
  }
}

// ---------------------------------------------------------------------------
// K8: out[n,i,:] = outacc[n,i,:] @ so3_w[L_OF[i]] (+ so3_b on i==0)
// Rows grouped by l so each 16-row WMMA tile shares one weight matrix.
// ---------------------------------------------------------------------------
__global__ void out_kernel(const float* __restrict__ outacc, const float* __restrict__ wp,
                           const float* __restrict__ bias, float* __restrict__ out, int N) {
  int tid = threadIdx.x, lane = tid & 31, wave = tid >> 5;
  int t0 = (N + 15) >> 4, t1 = (3 * N + 15) >> 4;
  int bt = blockIdx.x;
  int grp, rows, row0;
  const float* wl;
  if (bt < t0)            { grp = 0; rows = N;     row0 = bt * 16;             wl = wp; }
  else if (bt < t0 + t1)  { grp = 1; rows = 3 * N; row0 = (bt - t0) * 16;      wl = wp + 4096; }
  else                    { grp = 2; rows = 5 * N; row0 = (bt - t0 - t1) * 16; wl = wp + 8192; }
  int ar = row0 + (lane & 15);
  if (ar >= rows) ar = rows - 1;
  int an_, ai_;
  if (grp == 0)      { an_ = ar;     ai_ = 0; }
  else if (grp == 1) { an_ = ar / 3; ai_ = 1 + ar % 3; }
  else               { an_ = ar / 5; ai_ = 4 + ar % 5; }
  const float* arow = outacc + (size_t)an_ * 576 + ai_ * 64;
  v8f acc = {};
  for (int ks = 0; ks < 16; ++ks) {
    int kk = (ks << 2) + ((lane >> 4) << 1);
    v2f a;
    a.x = arow[kk];
    a.y = arow[kk + 1];
    acc = wmma_f32(a, frag_b_pk(wl, 16, wave, ks, lane), acc);
  }
  int col = (wave << 4) + (lane & 15), mb = (lane >> 4) << 3;
#pragma unroll
  for (int j = 0; j < 8; ++j) {
    int rr = row0 + mb + j;
    if (rr >= rows) continue;
    int n, i;
    if (grp == 0)      { n = rr;     i = 0; }
    else if (grp == 1) { n = rr / 3; i = 1 + rr % 3; }
    else               { n = rr / 5; i = 4 + rr % 5; }
    float v = acc[j];
    if (i == 0) v += bias[col];
    out[(size_t)n * 576 + i * 64 + col] = v;
  }
}

// ---------------------------------------------------------------------------
extern "C" void kernel_launch(void* const* d_in, const int* in_sizes, int n_in,
                              void* d_out, int out_size, void* d_ws, size_t ws_size,
                              hipStream_t stream) {
  const float* node_feats    = (const float*)d_in[0];
  const float* edge_embeds   = (const float*)d_in[1];
  const float* wigner        = (const float*)d_in[2];
  const float* wigner_inv    = (const float*)d_in[3];
  const float* to_grid       = (const float*)d_in[4];
  const float* from_grid     = (const float*)d_in[5];
  const float* species_embed = (const float*)d_in[6];
  const float* rw1 = (const float*)d_in[7];
  const float* rb1 = (const float*)d_in[8];
  const float* rw2 = (const float*)d_in[9];
  const float* rb2 = (const float*)d_in[10];
  const float* rw3 = (const float*)d_in[11];
  const float* rb3 = (const float*)d_in[12];
  const float* c1_w0 = (const float*)d_in[13];
  const float* c1_b0 = (const float*)d_in[14];
  const float* c1_w1 = (const float*)d_in[15];
  const float* c1_w2 = (const float*)d_in[16];
  const float* c2_w0 = (const float*)d_in[17];
  const float* c2_b0 = (const float*)d_in[18];
  const float* c2_w1 = (const float*)d_in[19];
  const float* c2_w2 = (const float*)d_in[20];
  const float* ln_scale  = (const float*)d_in[21];
  const float* ln_bias   = (const float*)d_in[22];
  const float* alpha_dot = (const float*)d_in[23];
  const float* so3_w     = (const float*)d_in[24];
  const float* so3_b     = (const float*)d_in[25];
  const int* node_species = (const int*)d_in[26];
  const int* senders      = (const int*)d_in[27];
  const int* receivers    = (const int*)d_in[28];
  const int N = in_sizes[26];
  const int E = in_sizes[27];
  float* out = (float*)d_out;

  // workspace carve-up (all fp32)
  float* ws = (float*)d_ws;
  size_t off = 0;
  float* msg1     = ws + off; off += (size_t)E * 1152;  // wigner-rotated messages
  float* rad      = ws + off; off += (size_t)E * 768;   // radial weights
  float* ybuf     = ws + off; off += (size_t)E * 576;   // conv1 output
  float* extrabuf = ws + off; off += (size_t)E * 576;   // scalar_alpha(512)+gate(64)
  float* m2in     = ws + off; off += (size_t)E * 576;   // gate/grid output
  float* m2out    = ws + off; off += (size_t)E * 576;   // conv2 output
  float* alphabuf = ws + off; off += (size_t)E * 8;
  float* amax     = ws + off; off += (size_t)N * 8;
  float* den      = ws + off; off += (size_t)N * 8;
  float* outacc   = ws + off; off += (size_t)N * 576;

  // packed weight buffers (fragment-major), repacked every call (~2.1 MB total)
  auto rp = [&](const float* W, int K, int Ncols) -> float* {
    float* dst = ws + off;
    off += (size_t)K * Ncols;
    int tot = (K / 4) * (Ncols / 16) * 32;
    repack_b_kernel<<<(tot + 255) / 256, 256, 0, stream>>>(W, dst, K, Ncols);
    return dst;
  };
  float* rw1p  = rp(rw1, 192, 64);
  float* rw2p  = rp(rw2, 64, 64);
  float* rw3p  = rp(rw3, 64, 768);
  float* c1w0p = rp(c1_w0, 384, 768);
  float* c1w1p = rp(c1_w1, 256, 256);
  float* c1w2p = rp(c1_w2, 128, 128);
  float* c2w0p = rp(c2_w0, 192, 192);
  float* c2w1p = rp(c2_w1, 128, 256);
  float* c2w2p = rp(c2_w2, 64, 128);
  float* so3p  = rp(so3_w, 64, 64);                 // l=0
  rp(so3_w + 4096, 64, 64);                         // l=1 (contiguous after so3p)
  rp(so3_w + 8192, 64, 64);                         // l=2
  (void)ws_size; (void)n_in; (void)out_size;

  const int et = (E + 15) / 16;

  init_kernel<<<(N * 576 + 255) / 256, 256, 0, stream>>>(amax, den, outacc, N);
  wigner_msg_kernel<<<E, 128, 0, stream>>>(node_feats, wigner, senders, receivers, msg1, E);
  radial_kernel<<<et, 128, 0, stream>>>(edge_embeds, species_embed, node_species, senders,
                                        receivers, rw1p, rb1, rw2p, rb2, rw3p, rb3, rad, E);
  so2conv_kernel<<<et, 128, 0, stream>>>(msg1, rad, c1w0p, c1_b0, c1w1p, c1w2p,
                                         ybuf, extrabuf, E, 128, 64, 576);
  grid_gate_kernel<<<(E * 64 + 255) / 256, 256, 0, stream>>>(ybuf, extrabuf, to_grid,
                                                             from_grid, m2in, E);
  so2conv_kernel<<<et, 128, 0, stream>>>(m2in, nullptr, c2w0p, c2_b0, c2w1p, c2w2p,
                                         m2out, nullptr, E, 64, 64, 0);
  alpha_kernel<<<(E + 7) / 8, 256, 0, stream>>>(extrabuf, ln_scale, ln_bias, alpha_dot,
                                                receivers, alphabuf, amax, E);
  exp_den_kernel<<<(E * 8 + 255) / 256, 256, 0, stream>>>(alphabuf, amax, den, receivers, E);
  combine_kernel<<<E, 192, 0, stream>>>(m2out, wigner_inv, alphabuf, den, receivers, outacc, E);
  int t0 = (N + 15) / 16, t1 = (3 * N + 15) / 16, t2 = (5 * N + 15) / 16;
  out_kernel<<<t0 + t1 + t2, 128, 0, stream>>>(outacc, so3p, so3_b, out, N);
}